// ChunkAttentionMaskLayer_20572893348529
// MI455X (gfx1250) — compile-verified
//
#include <hip/hip_runtime.h>
#include <stdint.h>

// ChunkAttentionMaskLayer: out[b,i,j] = 1 iff |i-j| <= N_SHIFT and
// mask[b, t] == 0 for all t strictly between min(i,j) and max(i,j)
// (inclusive of the far end). Each output row is a contiguous band
// [lo(i), hi(i)].  Pure store-bandwidth problem: 128 MiB of int32 out,
// ~5.8 us at 23.3 TB/s.  We stage 8x1024 tiles in LDS and blast them out
// with the CDNA5 Tensor Data Mover (TENSOR_STORE_FROM_LDS, TENSORcnt),
// two 16 KB DMA descriptors per tile.

#define L_DIM     1024
#define N_SHIFT_C 10
#define TILE_ROWS 8
#define THREADS   256

typedef uint32_t v4u __attribute__((ext_vector_type(4)));
typedef int32_t  v4i __attribute__((ext_vector_type(4)));
typedef int32_t  v8i __attribute__((ext_vector_type(8)));

#if defined(__gfx1250__) && __has_builtin(__builtin_amdgcn_tensor_store_from_lds)
#define USE_TDM 1
#else
#define USE_TDM 0
#endif

struct SmemT {
    int tile[TILE_ROWS][L_DIM];   // 32 KB; first member => LDS byte offset 0
    int lo[TILE_ROWS];
    int hi[TILE_ROWS];
};

#if USE_TDM
// Issue one TDM store: `rows` x 1024 int32 tile, LDS byte offset -> global.
__device__ __forceinline__ void tdm_store_tile(uint32_t lds_byte_off,
                                               uint64_t gaddr, int rows) {
    // D# group 0: count=1 | lds_addr | global_addr[56:0] | type=2
    v4u g0;
    g0.x = 1u;                                          // count=1, user mode
    g0.y = lds_byte_off;                                // LDS byte address
    g0.z = (uint32_t)(gaddr & 0xFFFFFFFFu);             // global_addr[31:0]
    g0.w = (uint32_t)((gaddr >> 32) & 0x1FFFFFFu)       // global_addr[56:32]
         | (2u << 30);                                  // type = 2 ("image")

    // D# group 1: data_size=4B, tensor 1024 x rows, tile 1024 x rows
    v8i g1;
    g1[0] = (2 << 16);               // workgroup_mask=0 | data_size=2 (4B)
    g1[1] = (L_DIM << 16);           // atomic_barrier=0 | tensor_dim0[15:0]=1024
    g1[2] = (rows << 16);            // tensor_dim0[31:16]=0 | tensor_dim1=rows
    g1[3] = (L_DIM << 16);           // tensor_dim1[31:16]=0 | tile_dim0=1024
    g1[4] = rows;                    // tile_dim1=rows | tile_dim2=0 (unused)
    g1[5] = L_DIM;                   // tensor_dim0_stride[31:0]=1024
    g1[6] = 0;                       // stride0[47:32]=0 | stride1[15:0]=0
    g1[7] = 0;                       // stride1[47:16]=0

    v4i gz4 = {0, 0, 0, 0};          // groups 2/3: dims 3/4 unused (2D tile)
    v8i gz8 = {0, 0, 0, 0, 0, 0, 0, 0};
    __builtin_amdgcn_tensor_store_from_lds(g0, g1, gz4, gz4, gz8, 0);
}
#endif

__global__ __launch_bounds__(THREADS)
void band_mask_kernel(const int* __restrict__ mask, int* __restrict__ out) {
    __shared__ SmemT sm;
    const int tid  = threadIdx.x;
    const int row0 = blockIdx.x * TILE_ROWS;        // global row = b*L + i

    // ---- 1) per-row band bounds (threads 0..7, one row each) ----
    if (tid < TILE_ROWS) {
        const int grow = row0 + tid;
        const int b = grow >> 10;
        const int i = grow & (L_DIM - 1);
        const int* mrow = mask + b * L_DIM;
        int lo = i;
        for (int t = i; t > 0 && t > i - N_SHIFT_C; --t) {
            if (mrow[t] != 0) break;
            lo = t - 1;
        }
        int hi = i;
        for (int t = i + 1; t < L_DIM && t <= i + N_SHIFT_C; ++t) {
            if (mrow[t] != 0) break;
            hi = t;
        }
        sm.lo[tid] = lo;
        sm.hi[tid] = hi;
    }
    __syncthreads();

    const int r  = tid >> 5;                        // 32 threads per row
    const int c0 = (tid & 31) * 32;                 // 32 columns per thread
    const int lo = sm.lo[r];
    const int hi = sm.hi[r];

#if USE_TDM
    // ---- 2) build the tile in LDS (b128 DS stores) ----
    #pragma unroll
    for (int k = 0; k < 8; ++k) {
        const int c = c0 + k * 4;
        v4i v;
        v.x = (c     >= lo && c     <= hi) ? 1 : 0;
        v.y = (c + 1 >= lo && c + 1 <= hi) ? 1 : 0;
        v.z = (c + 2 >= lo && c + 2 <= hi) ? 1 : 0;
        v.w = (c + 3 >= lo && c + 3 <= hi) ? 1 : 0;
        *(v4i*)&sm.tile[r][c] = v;
    }
    __syncthreads();   // DS writes visible before TDM reads LDS

    // ---- 3) two TDM stores drain the 32 KB tile to HBM ----
    if (tid == 0) {
        const int half = TILE_ROWS / 2;                         // 4 rows
        const uint32_t half_bytes = half * L_DIM * sizeof(int); // 16 KB
        const uint64_t gbase = (uint64_t)(uintptr_t)(out + (size_t)row0 * L_DIM);
        tdm_store_tile(0u,         gbase,                         half);
        tdm_store_tile(half_bytes, gbase + (uint64_t)half_bytes,  half);
        // LDS must stay alive until the DMA has read it.
        __builtin_amdgcn_s_wait_tensorcnt(0);
    }
#else
    // ---- fallback: direct wide global stores ----
    int* orow = out + (size_t)(row0 + r) * L_DIM;
    #pragma unroll
    for (int k = 0; k < 8; ++k) {
        const int c = c0 + k * 4;
        v4i v;
        v.x = (c     >= lo && c     <= hi) ? 1 : 0;
        v.y = (c + 1 >= lo && c + 1 <= hi) ? 1 : 0;
        v.z = (c + 2 >= lo && c + 2 <= hi) ? 1 : 0;
        v.w = (c + 3 >= lo && c + 3 <= hi) ? 1 : 0;
        *(v4i*)&orow[c] = v;
    }
#endif
}

extern "C" void kernel_launch(void* const* d_in, const int* in_sizes, int n_in,
                              void* d_out, int out_size, void* d_ws, size_t ws_size,
                              hipStream_t stream) {
    (void)in_sizes; (void)n_in; (void)d_ws; (void)ws_size; (void)out_size;
    const int* mask = (const int*)d_in[0];      // (32, 1024) int32
    int* out        = (int*)d_out;              // (32, 1024, 1024) int32
    const int total_rows = 32 * 1024;
    const int grid = total_rows / TILE_ROWS;    // 4096 workgroups
    band_mask_kernel<<<grid, THREADS, 0, stream>>>(mask, out);
}